// GCNJumpingKnowledge2_515396076079
// MI455X (gfx1250) — compile-verified
//
#include <hip/hip_runtime.h>
#include <hip/hip_bf16.h>
#include <math.h>

#define FEAT 128
#define ODIM 10

typedef __attribute__((ext_vector_type(2))) float v2f;
typedef __attribute__((ext_vector_type(8))) float v8f;

#if __has_builtin(__builtin_amdgcn_wmma_f32_16x16x4_f32)
#define HAVE_WMMA_F32 1
#else
#define HAVE_WMMA_F32 0
#endif

// ---------------- utility kernels ----------------

__global__ void zero_i32(int* __restrict__ p, int n) {
    int i = blockIdx.x * blockDim.x + threadIdx.x;
    if (i < n) p[i] = 0;
}

__global__ void count_deg(const int* __restrict__ dst, int* __restrict__ deg, int nE) {
    int e = blockIdx.x * blockDim.x + threadIdx.x;
    if (e < nE) atomicAdd(&deg[dst[e]], 1);
}

__global__ void deg_finalize(const int* __restrict__ deg, float* __restrict__ dsq,
                             float* __restrict__ invd, int n) {
    int i = blockIdx.x * blockDim.x + threadIdx.x;
    if (i < n) {
        float d = (float)deg[i] + 1.0f;   // self-loop
        dsq[i]  = rsqrtf(d);
        invd[i] = 1.0f / d;
    }
}

// single-workgroup chunked Hillis-Steele scan: rowptr[i+1] = sum(deg[0..i]), rowptr[0]=0
__global__ __launch_bounds__(1024) void scan_rowptr(const int* __restrict__ deg,
                                                    int* __restrict__ rowptr, int n) {
    __shared__ int buf[1024];
    __shared__ int carry;
    int tid = threadIdx.x;
    if (tid == 0) carry = 0;
    __syncthreads();
    for (int base = 0; base < n; base += 1024) {
        int i = base + tid;
        int v = (i < n) ? deg[i] : 0;
        buf[tid] = v;
        __syncthreads();
        for (int off = 1; off < 1024; off <<= 1) {
            int t = (tid >= off) ? buf[tid - off] : 0;
            __syncthreads();
            buf[tid] += t;
            __syncthreads();
        }
        int inc   = buf[tid];     // inclusive
        int total = buf[1023];
        int c     = carry;
        if (i < n) rowptr[i + 1] = c + inc;
        __syncthreads();
        if (tid == 0) carry = c + total;
        __syncthreads();
    }
    if (tid == 0) rowptr[0] = 0;
}

__global__ void fill_csr(const int* __restrict__ dst, const int* __restrict__ rowptr,
                         int* __restrict__ fill, int* __restrict__ eid, int nE) {
    int e = blockIdx.x * blockDim.x + threadIdx.x;
    if (e < nE) {
        int d = dst[e];
        int pos = atomicAdd(&fill[d], 1);
        eid[rowptr[d] + pos] = e;
    }
}

// per-bucket insertion sort (avg degree ~32) -> bit-deterministic accumulation order
__global__ void sort_buckets(int* __restrict__ eid, const int* __restrict__ rowptr, int n) {
    int node = blockIdx.x * blockDim.x + threadIdx.x;
    if (node >= n) return;
    int beg = rowptr[node], end = rowptr[node + 1];
    for (int i = beg + 1; i < end; ++i) {
        int key = eid[i];
        int j = i - 1;
        while (j >= beg && eid[j] > key) { eid[j + 1] = eid[j]; --j; }
        eid[j + 1] = key;
    }
}

// ---------------- WMMA GEMM: H = X @ W  (X:[N,128], W:[128,128]) ----------------
// block = 256 threads = 8 waves; each wave computes 16 rows x 128 cols.
// W staged in LDS K-pair-interleaved: Wl[k2*128+n] = {W[2k2][n], W[2k2+1][n]}
// so every B fragment is a single aligned ds_load_b64 into an even VGPR pair.

__global__ __launch_bounds__(256) void gemm128_wmma(const float* __restrict__ X,
                                                    const float* __restrict__ W,
                                                    float* __restrict__ H, int nrows) {
    __shared__ v2f Wl[64 * FEAT];   // 64 KB
    int tid = threadIdx.x;

    for (int idx = tid; idx < 64 * FEAT; idx += 256) {
        int k2 = idx >> 7;
        int n  = idx & (FEAT - 1);
        v2f p;
        p.x = W[(2 * k2) * FEAT + n];       // coalesced over n
        p.y = W[(2 * k2 + 1) * FEAT + n];
        Wl[idx] = p;
    }
    __syncthreads();

    int wave = tid >> 5;
    int lane = tid & 31;
    int half = lane >> 4;   // 0: lanes 0-15, 1: lanes 16-31
    int l16  = lane & 15;

    int row0 = blockIdx.x * 128 + wave * 16;

#if HAVE_WMMA_F32
    v8f acc[8] = {};
    for (int k0 = 0; k0 < FEAT; k0 += 4) {
        int ar = row0 + l16;
        if (ar >= nrows) ar = nrows - 1;     // clamp: EXEC must stay all-ones for WMMA
        // A fragment: {X[r][k0+2h], X[r][k0+2h+1]} -> one global_load_b64
        v2f a = *(const v2f*)(X + (size_t)ar * FEAT + k0 + 2 * half);
        int kp = (k0 >> 1) + half;           // interleaved pair row
#pragma unroll
        for (int j = 0; j < 8; ++j) {
            v2f b = Wl[kp * FEAT + j * 16 + l16];   // one ds_load_b64
            acc[j] = __builtin_amdgcn_wmma_f32_16x16x4_f32(
                false, a, false, b, (short)0, acc[j], false, false);
        }
    }
#pragma unroll
    for (int j = 0; j < 8; ++j) {
        int c = j * 16 + l16;
#pragma unroll
        for (int i = 0; i < 8; ++i) {
            int r = row0 + i + 8 * half;
            if (r < nrows) H[(size_t)r * FEAT + c] = acc[j][i];
        }
    }
#else
    int base = blockIdx.x * 128;
    for (int idx = tid; idx < 128 * FEAT; idx += 256) {
        int r = base + (idx >> 7);
        int c = idx & (FEAT - 1);
        if (r < nrows) {
            float s = 0.0f;
            const float* xr = X + (size_t)r * FEAT;
            for (int k = 0; k < FEAT; ++k) {
                v2f p = Wl[(k >> 1) * FEAT + c];
                s = fmaf(xr[k], (k & 1) ? p.y : p.x, s);
            }
            H[(size_t)r * FEAT + c] = s;
        }
    }
#endif
}

// ---------------- fused gather aggregation + self-loop + bias + relu ----------------
// one wave per dst node; lane handles 4 consecutive features; accumulate in registers.
// OUT[n] = relu( sum_{e: dst=n} H[src_e]*dsq[src_e]*dsq[n] + H[n]*invd[n] + b )

__global__ __launch_bounds__(256) void gather_layer(const float* __restrict__ H,
                                                    const int* __restrict__ srcArr,
                                                    const int* __restrict__ eid,
                                                    const int* __restrict__ rowptr,
                                                    const float* __restrict__ dsq,
                                                    const float* __restrict__ invd,
                                                    const float* __restrict__ b,
                                                    float* __restrict__ OUT, int n) {
    int lane = threadIdx.x & 31;
    int node = blockIdx.x * (blockDim.x >> 5) + (threadIdx.x >> 5);
    if (node >= n) return;

    int beg = rowptr[node];
    int end = rowptr[node + 1];
    float dn = dsq[node];

    float4 acc = make_float4(0.f, 0.f, 0.f, 0.f);
    for (int p = beg; p < end; ++p) {
        int e = eid[p];
        int s = srcArr[e];
        float w = dn * dsq[s];
        float4 v = *(const float4*)(H + (size_t)s * FEAT + lane * 4);  // coalesced 512B row
        acc.x = fmaf(v.x, w, acc.x);
        acc.y = fmaf(v.y, w, acc.y);
        acc.z = fmaf(v.z, w, acc.z);
        acc.w = fmaf(v.w, w, acc.w);
    }
    float id = invd[node];
    float4 hv = *(const float4*)(H + (size_t)node * FEAT + lane * 4);
    float4 bv = *(const float4*)(b + lane * 4);
    float4 o;
    o.x = fmaxf(fmaf(hv.x, id, acc.x) + bv.x, 0.f);
    o.y = fmaxf(fmaf(hv.y, id, acc.y) + bv.y, 0.f);
    o.z = fmaxf(fmaf(hv.z, id, acc.z) + bv.z, 0.f);
    o.w = fmaxf(fmaf(hv.w, id, acc.w) + bv.w, 0.f);
    *(float4*)(OUT + (size_t)node * FEAT + lane * 4) = o;
}

// ---------------- readout: softmax(concat(h1,h2,h3) @ Wr + br) ----------------

__global__ __launch_bounds__(256) void readout(const float* __restrict__ H1,
                                               const float* __restrict__ H2,
                                               const float* __restrict__ H3,
                                               const float* __restrict__ Wr,
                                               const float* __restrict__ br,
                                               float* __restrict__ OUT, int n) {
    int lane   = threadIdx.x & 31;
    int warpId = blockIdx.x * (blockDim.x >> 5) + (threadIdx.x >> 5);
    if (warpId >= n) return;

    const float* r1 = H1 + (size_t)warpId * FEAT;
    const float* r2 = H2 + (size_t)warpId * FEAT;
    const float* r3 = H3 + (size_t)warpId * FEAT;

    float acc[ODIM];
#pragma unroll
    for (int o = 0; o < ODIM; ++o) acc[o] = 0.0f;

    for (int f = lane; f < 3 * FEAT; f += 32) {   // branch uniform per iteration
        float xv = (f < FEAT) ? r1[f] : (f < 2 * FEAT ? r2[f - FEAT] : r3[f - 2 * FEAT]);
        const float* wr = Wr + f * ODIM;
#pragma unroll
        for (int o = 0; o < ODIM; ++o) acc[o] = fmaf(xv, wr[o], acc[o]);
    }
#pragma unroll
    for (int o = 0; o < ODIM; ++o)
        for (int off = 16; off > 0; off >>= 1)
            acc[o] += __shfl_xor(acc[o], off, 32);

    if (lane == 0) {
        float m = -1e30f;
#pragma unroll
        for (int o = 0; o < ODIM; ++o) {
            acc[o] += br[o];
            m = fmaxf(m, acc[o]);
        }
        float ssum = 0.0f;
#pragma unroll
        for (int o = 0; o < ODIM; ++o) {
            acc[o] = expf(acc[o] - m);
            ssum += acc[o];
        }
        float inv = 1.0f / ssum;
#pragma unroll
        for (int o = 0; o < ODIM; ++o) OUT[(size_t)warpId * ODIM + o] = acc[o] * inv;
    }
}

// ---------------- launcher ----------------

extern "C" void kernel_launch(void* const* d_in, const int* in_sizes, int n_in,
                              void* d_out, int out_size, void* d_ws, size_t ws_size,
                              hipStream_t stream) {
    const float* x    = (const float*)d_in[0];
    const int*   edge = (const int*)d_in[1];
    const float* W0   = (const float*)d_in[2];
    const float* b0   = (const float*)d_in[3];
    const float* W1   = (const float*)d_in[4];
    const float* b1   = (const float*)d_in[5];
    const float* W2   = (const float*)d_in[6];
    const float* b2   = (const float*)d_in[7];
    const float* Wr   = (const float*)d_in[8];
    const float* br   = (const float*)d_in[9];
    float* out = (float*)d_out;

    int N = in_sizes[0] / FEAT;
    int E = in_sizes[1] / 2;
    const int* src = edge;
    const int* dst = edge + E;

    size_t nf = (size_t)N * FEAT;
    char* ws = (char*)d_ws;
    float* T      = (float*)ws; ws += nf * sizeof(float);
    float* H1     = (float*)ws; ws += nf * sizeof(float);
    float* H2     = (float*)ws; ws += nf * sizeof(float);
    float* H3     = (float*)ws; ws += nf * sizeof(float);
    float* dsq    = (float*)ws; ws += (size_t)N * sizeof(float);
    float* invd   = (float*)ws; ws += (size_t)N * sizeof(float);
    int*   deg    = (int*)ws;   ws += (size_t)N * sizeof(int);
    int*   rowptr = (int*)ws;   ws += (size_t)(N + 1) * sizeof(int);
    int*   fill   = (int*)ws;   ws += (size_t)N * sizeof(int);
    int*   eid    = (int*)ws;   ws += (size_t)E * sizeof(int);

    // ---- build degree tables + CSR-by-dst (reused across all 3 layers) ----
    zero_i32<<<(N + 255) / 256, 256, 0, stream>>>(deg, N);
    count_deg<<<(E + 255) / 256, 256, 0, stream>>>(dst, deg, E);
    deg_finalize<<<(N + 255) / 256, 256, 0, stream>>>(deg, dsq, invd, N);
    scan_rowptr<<<1, 1024, 0, stream>>>(deg, rowptr, N);
    zero_i32<<<(N + 255) / 256, 256, 0, stream>>>(fill, N);
    fill_csr<<<(E + 255) / 256, 256, 0, stream>>>(dst, rowptr, fill, eid, E);
    sort_buckets<<<(N + 255) / 256, 256, 0, stream>>>(eid, rowptr, N);

    // ---- 3 GCN layers: WMMA GEMM + fused gather/self-loop/bias/relu ----
    const float* Ws[3] = {W0, W1, W2};
    const float* bs[3] = {b0, b1, b2};
    float*       Hs[3] = {H1, H2, H3};
    const float* Xin = x;

    for (int l = 0; l < 3; ++l) {
        gemm128_wmma<<<(N + 127) / 128, 256, 0, stream>>>(Xin, Ws[l], T, N);
        gather_layer<<<(N + 7) / 8, 256, 0, stream>>>(T, src, eid, rowptr, dsq, invd,
                                                      bs[l], Hs[l], N);
        Xin = Hs[l];
    }

    readout<<<(N + 7) / 8, 256, 0, stream>>>(H1, H2, H3, Wr, br, out, N);
}